// TagHAN_15899968930389
// MI455X (gfx1250) — compile-verified
//
#include <hip/hip_runtime.h>
#include <cstdint>

// ---------------- problem constants (from reference) ----------------
constexpr int NHEAD  = 4;
constexpr int DHEAD  = 32;
constexpr int HIDDEN = NHEAD * DHEAD;   // 128
constexpr int NV     = 100000;
constexpr int NT     = 30000;
constexpr int NVD    = 50000;
constexpr int NTD    = 15000;
constexpr int ETV    = 800000;
constexpr int EVV    = 800000;
constexpr int EVT    = 300000;
constexpr int VID_IN = 1024;
constexpr int TAG_IN = 256;
constexpr float NEG_SLOPE = 0.2f;

typedef __attribute__((ext_vector_type(2))) float v2f;
typedef __attribute__((ext_vector_type(8))) float v8f;

// ---------------- helpers ----------------
__device__ __forceinline__ void atomicMaxFloat(float* addr, float val) {
    // sign-aware monotone trick: ints order like floats for >=0,
    // unsigned-min orders correctly for negatives. Init must be -inf bits.
    if (val >= 0.0f) atomicMax((int*)addr, __float_as_int(val));
    else             atomicMin((unsigned int*)addr, __float_as_uint(val));
}

__global__ void fill_kernel(float* __restrict__ p, float v, long long n) {
    long long i = (long long)blockIdx.x * blockDim.x + threadIdx.x;
    long long stride = (long long)gridDim.x * blockDim.x;
    for (; i < n; i += stride) p[i] = v;
}

__global__ void copy_kernel(const float* __restrict__ in, float* __restrict__ out, int n) {
    int t = blockIdx.x * blockDim.x + threadIdx.x;
    if (t < n) out[t] = in[t];
}

// ---------------- BN fold: W'[k,n] = scale[k]*W[k,n]; b'[n] = bias[n] + sum_k (beta[k]-mean[k]*scale[k])*W[k,n]
__global__ __launch_bounds__(256)
void bn_fold_kernel(const float* __restrict__ g, const float* __restrict__ b,
                    const float* __restrict__ m, const float* __restrict__ v,
                    const float* __restrict__ W, const float* __restrict__ bias,
                    float* __restrict__ Wf, float* __restrict__ bf, int K) {
    __shared__ float red[256];
    const int n = blockIdx.x;               // 0..127
    float s = 0.0f;
    for (int k = threadIdx.x; k < K; k += 256) {
        float sc = g[k] * rsqrtf(v[k] + 1e-5f);
        float w  = W[(size_t)k * HIDDEN + n];
        Wf[(size_t)k * HIDDEN + n] = sc * w;
        s += (b[k] - m[k] * sc) * w;
    }
    red[threadIdx.x] = s;
    __syncthreads();
    for (int off = 128; off > 0; off >>= 1) {
        if (threadIdx.x < off) red[threadIdx.x] += red[threadIdx.x + off];
        __syncthreads();
    }
    if (threadIdx.x == 0) bf[n] = bias[n] + red[0];
}

// ---------------- tag_in = tag_feat + tag_embed[tag_ids]
__global__ void tag_input_kernel(const float* __restrict__ tag_feat,
                                 const float* __restrict__ tag_embed,
                                 const int* __restrict__ tag_ids,
                                 float* __restrict__ out, int total) {
    int t = blockIdx.x * blockDim.x + threadIdx.x;
    if (t >= total) return;
    int i = t >> 8;          // row (TAG_IN = 256)
    int k = t & 255;
    out[t] = tag_feat[t] + tag_embed[(size_t)tag_ids[i] * TAG_IN + k];
}

// ---------------- fp32 WMMA GEMM: C[M,128] = A[M,K] @ W[K,128] (+bias) (+ relu(hm+gbias))
// 8 waves/block; wave w owns output tile rows [16*bx .. +16), cols [16*w .. +16)
// A frag (16x4 f32): lanes 0-15 -> K=k,k+1 ; lanes 16-31 -> K=k+2,k+3 (per ISA layout)
// B frag (4x16 f32): VGPR0 = rows k / k+2 ; VGPR1 = rows k+1 / k+3
// C/D (16x16 f32): VGPR r -> row r (lanes 0-15) / row r+8 (lanes 16-31)
__global__ __launch_bounds__(256)
void gemm128_wmma(const float* __restrict__ A, const float* __restrict__ W,
                  const float* __restrict__ bias,
                  const float* __restrict__ hm, const float* __restrict__ gbias,
                  float* __restrict__ C, int M, int K) {
    const int lane = threadIdx.x & 31;
    const int wave = threadIdx.x >> 5;          // 0..7 -> column tile
    const int n0   = wave * 16;
    const int l16  = lane & 15;
    const int hi   = lane >> 4;                 // 0 or 1
    const int m0   = blockIdx.x * 16;

    int m  = m0 + l16;
    int mc = m < M ? m : M - 1;                 // clamp OOB rows; stores are guarded
    const int n = n0 + l16;

    const float* Arow = A + (size_t)mc * K;
    v8f acc = {0.f, 0.f, 0.f, 0.f, 0.f, 0.f, 0.f, 0.f};

    #pragma unroll 2
    for (int kk = 0; kk < K; kk += 4) {
        const int ka = kk + hi * 2;
        float2 av = *reinterpret_cast<const float2*>(Arow + ka);
        v2f a; a[0] = av.x; a[1] = av.y;
        v2f b;
        b[0] = W[(size_t)ka * HIDDEN + n];
        b[1] = W[(size_t)(ka + 1) * HIDDEN + n];
        acc = __builtin_amdgcn_wmma_f32_16x16x4_f32(
            false, a, false, b, (short)0, acc, false, false);
    }

    const bool has_bias = (bias != nullptr);
    const bool has_hm   = (hm != nullptr);
    #pragma unroll
    for (int r = 0; r < 8; ++r) {
        int row = m0 + r + hi * 8;
        if (row < M) {
            float val = acc[r];
            if (has_bias) val += bias[n];
            if (has_hm) {
                float t = hm[(size_t)row * HIDDEN + n] + gbias[n];
                val += (t > 0.0f) ? t : 0.0f;
            }
            C[(size_t)row * HIDDEN + n] = val;
        }
    }
}

// ---------------- el/er: out[i*4+h] = sum_d feat[i,h*32+d] * attn[h*32+d]
__global__ void attn_dot_kernel(const float* __restrict__ feat,
                                const float* __restrict__ attn,
                                float* __restrict__ out, int total) {
    int t = blockIdx.x * blockDim.x + threadIdx.x;
    if (t >= total) return;
    int node = t >> 2, h = t & 3;
    const float* f = feat + (size_t)node * HIDDEN + h * DHEAD;
    const float* a = attn + h * DHEAD;
    float s = 0.0f;
    #pragma unroll
    for (int d = 0; d < DHEAD; ++d) s += f[d] * a[d];
    out[t] = s;
}

// ---------------- edge pass 1: e = leaky_relu(el[src]+er[dst]); segment max
__global__ void edge_logit_kernel(const int* __restrict__ src, const int* __restrict__ dst,
                                  const float* __restrict__ el, const float* __restrict__ er,
                                  float* __restrict__ e, float* __restrict__ mx, int total) {
    int t = blockIdx.x * blockDim.x + threadIdx.x;
    if (t >= total) return;
    int eid = t >> 2, h = t & 3;
    int s = src[eid], d = dst[eid];
    float v = el[s * NHEAD + h] + er[d * NHEAD + h];
    v = (v >= 0.0f) ? v : NEG_SLOPE * v;
    e[t] = v;
    atomicMaxFloat(&mx[d * NHEAD + h], v);
}

// ---------------- edge pass 2: ex = exp(e - mx[dst]); segment sum
__global__ void edge_exp_kernel(const int* __restrict__ dst, const float* __restrict__ mx,
                                float* __restrict__ e, float* __restrict__ sm, int total) {
    int t = blockIdx.x * blockDim.x + threadIdx.x;
    if (t >= total) return;
    int eid = t >> 2, h = t & 3;
    int d = dst[eid];
    float ex = expf(e[t] - mx[d * NHEAD + h]);
    e[t] = ex;
    atomicAdd(&sm[d * NHEAD + h], ex);
}

// ---------------- edge pass 3: hm[dst] += (ex/sm[dst]) * feat_src[src]  (one wave per edge)
__global__ __launch_bounds__(256)
void edge_scatter_kernel(const int* __restrict__ src, const int* __restrict__ dst,
                         const float* __restrict__ e, const float* __restrict__ sm,
                         const float* __restrict__ feat_src, float* __restrict__ hm, int E) {
    int gw = (int)((blockIdx.x * (unsigned)blockDim.x + threadIdx.x) >> 5);
    if (gw >= E) return;
    int lane = threadIdx.x & 31;
    int s = src[gw], d = dst[gw];
    #pragma unroll
    for (int h = 0; h < NHEAD; ++h) {
        float a = e[gw * NHEAD + h] / sm[d * NHEAD + h];
        int f = h * DHEAD + lane;
        atomicAdd(&hm[(size_t)d * HIDDEN + f], a * feat_src[(size_t)s * HIDDEN + f]);
    }
}

// ---------------- semantic attention: acc += sum_i dot(tanh(z[i]@W1+b1), w2)
__global__ __launch_bounds__(128)
void sematt_kernel(const float* __restrict__ z, const float* __restrict__ w1,
                   const float* __restrict__ b1, const float* __restrict__ w2,
                   float* __restrict__ acc) {
    __shared__ float row[128];
    __shared__ float red[128];
    const int i = blockIdx.x;
    const int j = threadIdx.x;
    row[j] = z[(size_t)i * HIDDEN + j];
    __syncthreads();
    float s = b1[j];
    for (int k = 0; k < HIDDEN; ++k) s += row[k] * w1[k * HIDDEN + j];
    s = tanhf(s) * w2[j];
    red[j] = s;
    __syncthreads();
    for (int off = 64; off > 0; off >>= 1) {
        if (j < off) red[j] += red[j + off];
        __syncthreads();
    }
    if (j == 0) atomicAdd(acc, red[0]);
}

__global__ void beta_kernel(const float* __restrict__ wacc, float* __restrict__ beta, float invM) {
    float w0 = wacc[0] * invM, w1 = wacc[1] * invM;
    float mx = fmaxf(w0, w1);
    float e0 = expf(w0 - mx), e1 = expf(w1 - mx);
    float s = e0 + e1;
    beta[0] = e0 / s;
    beta[1] = e1 / s;
}

// ---------------- final: out = l2norm(beta0*a + beta1*b)
__global__ __launch_bounds__(128)
void final_video_kernel(const float* __restrict__ a, const float* __restrict__ b,
                        const float* __restrict__ beta, float* __restrict__ out) {
    __shared__ float red[128];
    const int i = blockIdx.x;
    const int j = threadIdx.x;
    float h = beta[0] * a[(size_t)i * HIDDEN + j] + beta[1] * b[(size_t)i * HIDDEN + j];
    red[j] = h * h;
    __syncthreads();
    for (int off = 64; off > 0; off >>= 1) {
        if (j < off) red[j] += red[j + off];
        __syncthreads();
    }
    float nrm = sqrtf(red[0]);
    out[(size_t)i * HIDDEN + j] = h / fmaxf(nrm, 1e-12f);
}

// ================= host side =================
extern "C" void kernel_launch(void* const* d_in, const int* in_sizes, int n_in,
                              void* d_out, int out_size, void* d_ws, size_t ws_size,
                              hipStream_t stream) {
    (void)in_sizes; (void)n_in; (void)out_size; (void)ws_size;

    const float* video_feat = (const float*)d_in[0];
    const float* tag_feat   = (const float*)d_in[1];
    const float* v_g        = (const float*)d_in[2];
    const float* v_b        = (const float*)d_in[3];
    const float* v_m        = (const float*)d_in[4];
    const float* v_v        = (const float*)d_in[5];
    const float* video_w    = (const float*)d_in[6];
    const float* video_b    = (const float*)d_in[7];
    const float* tag_embed  = (const float*)d_in[8];
    const float* t_g        = (const float*)d_in[9];
    const float* t_b        = (const float*)d_in[10];
    const float* t_m        = (const float*)d_in[11];
    const float* t_v        = (const float*)d_in[12];
    const float* tag_w      = (const float*)d_in[13];
    const float* tag_b      = (const float*)d_in[14];
    const float* gat_fc_w   = (const float*)d_in[15];   // [3,128,128]
    const float* gat_attn_l = (const float*)d_in[16];   // [3,128]
    const float* gat_attn_r = (const float*)d_in[17];   // [3,128]
    const float* gat_bias   = (const float*)d_in[18];   // [3,128]
    const float* gat_upd_w  = (const float*)d_in[19];   // [3,128,128]
    const float* gat_upd_b  = (const float*)d_in[20];   // [3,128]
    const float* sa_w1      = (const float*)d_in[21];
    const float* sa_b1      = (const float*)d_in[22];
    const float* sa_w2      = (const float*)d_in[23];
    const int*   tv_src     = (const int*)d_in[24];
    const int*   tv_dst     = (const int*)d_in[25];
    const int*   vv_src     = (const int*)d_in[26];
    const int*   vv_dst     = (const int*)d_in[27];
    const int*   vt_src     = (const int*)d_in[28];
    const int*   vt_dst     = (const int*)d_in[29];
    const int*   tag_ids    = (const int*)d_in[30];

    float* out = (float*)d_out;

    // -------- workspace carve-out (floats) --------
    float* ws = (float*)d_ws;
    size_t off = 0;
    auto carve = [&](size_t n) { float* p = ws + off; off += n; return p; };
    float* h_v      = carve((size_t)NV * HIDDEN);
    float* h_t      = carve((size_t)NT * HIDDEN);
    float* out_tv   = carve((size_t)NVD * HIDDEN);
    float* out_vv   = carve((size_t)NVD * HIDDEN);
    float* out_vt   = carve((size_t)NTD * HIDDEN);
    float* wfold_v  = carve((size_t)VID_IN * HIDDEN);
    float* bfold_v  = carve(HIDDEN);
    float* wfold_t  = carve((size_t)TAG_IN * HIDDEN);
    float* bfold_t  = carve(HIDDEN);
    float* tag_in   = carve((size_t)NT * TAG_IN);
    float* feat_src = carve((size_t)NV * HIDDEN);
    float* feat_dst = carve((size_t)NVD * HIDDEN);
    float* el_buf   = carve((size_t)NV * NHEAD);
    float* er_buf   = carve((size_t)NVD * NHEAD);
    float* e_edge   = carve((size_t)ETV * NHEAD);
    float* mx_buf   = carve((size_t)NVD * NHEAD);
    float* sm_buf   = carve((size_t)NVD * NHEAD);
    float* hm_buf   = carve((size_t)NVD * HIDDEN);
    float* wacc     = carve(2);
    float* beta     = carve(2);

    const float NEG_INF = -__builtin_inff();

    // -------- input projections --------
    bn_fold_kernel<<<HIDDEN, 256, 0, stream>>>(v_g, v_b, v_m, v_v, video_w, video_b,
                                               wfold_v, bfold_v, VID_IN);
    gemm128_wmma<<<NV / 16, 256, 0, stream>>>(video_feat, wfold_v, bfold_v,
                                              nullptr, nullptr, h_v, NV, VID_IN);

    tag_input_kernel<<<(NT * TAG_IN + 255) / 256, 256, 0, stream>>>(
        tag_feat, tag_embed, tag_ids, tag_in, NT * TAG_IN);
    bn_fold_kernel<<<HIDDEN, 256, 0, stream>>>(t_g, t_b, t_m, t_v, tag_w, tag_b,
                                               wfold_t, bfold_t, TAG_IN);
    gemm128_wmma<<<NT / 16, 256, 0, stream>>>(tag_in, wfold_t, bfold_t,
                                              nullptr, nullptr, h_t, NT, TAG_IN);

    // -------- three GAT metapaths --------
    struct MP {
        const float* hsrc; int Ns;
        const float* hdst; int nd;
        const int* src; const int* dst; int E;
        float* outp;
    } mp[3] = {
        { h_t, NT,  h_v, NVD, tv_src, tv_dst, ETV, out_tv },
        { h_v, NV,  h_v, NVD, vv_src, vv_dst, EVV, out_vv },
        { h_v, NV,  h_t, NTD, vt_src, vt_dst, EVT, out_vt },
    };

    for (int p = 0; p < 3; ++p) {
        const float* fcw = gat_fc_w   + (size_t)p * HIDDEN * HIDDEN;
        const float* al  = gat_attn_l + (size_t)p * HIDDEN;
        const float* ar  = gat_attn_r + (size_t)p * HIDDEN;
        const float* gb  = gat_bias   + (size_t)p * HIDDEN;
        const float* uw  = gat_upd_w  + (size_t)p * HIDDEN * HIDDEN;
        const float* ub  = gat_upd_b  + (size_t)p * HIDDEN;
        const int Ns = mp[p].Ns, nd = mp[p].nd, E = mp[p].E;

        gemm128_wmma<<<(Ns + 15) / 16, 256, 0, stream>>>(mp[p].hsrc, fcw, nullptr,
                                                         nullptr, nullptr, feat_src, Ns, HIDDEN);
        gemm128_wmma<<<(nd + 15) / 16, 256, 0, stream>>>(mp[p].hdst, fcw, nullptr,
                                                         nullptr, nullptr, feat_dst, nd, HIDDEN);

        attn_dot_kernel<<<(Ns * NHEAD + 255) / 256, 256, 0, stream>>>(feat_src, al, el_buf, Ns * NHEAD);
        attn_dot_kernel<<<(nd * NHEAD + 255) / 256, 256, 0, stream>>>(feat_dst, ar, er_buf, nd * NHEAD);

        fill_kernel<<<256, 256, 0, stream>>>(mx_buf, NEG_INF, (long long)nd * NHEAD);
        fill_kernel<<<256, 256, 0, stream>>>(sm_buf, 0.0f, (long long)nd * NHEAD);
        fill_kernel<<<4096, 256, 0, stream>>>(hm_buf, 0.0f, (long long)nd * HIDDEN);

        edge_logit_kernel<<<(E * NHEAD + 255) / 256, 256, 0, stream>>>(
            mp[p].src, mp[p].dst, el_buf, er_buf, e_edge, mx_buf, E * NHEAD);
        edge_exp_kernel<<<(E * NHEAD + 255) / 256, 256, 0, stream>>>(
            mp[p].dst, mx_buf, e_edge, sm_buf, E * NHEAD);
        edge_scatter_kernel<<<(E + 7) / 8, 256, 0, stream>>>(
            mp[p].src, mp[p].dst, e_edge, sm_buf, feat_src, hm_buf, E);

        // out = h0 @ upd_w + upd_b + relu(hm + gat_bias)
        gemm128_wmma<<<(nd + 15) / 16, 256, 0, stream>>>(mp[p].hdst, uw, ub,
                                                         hm_buf, gb, mp[p].outp, nd, HIDDEN);
    }

    // -------- semantic attention (video: 2 paths; tag: identity) --------
    fill_kernel<<<1, 32, 0, stream>>>(wacc, 0.0f, 2);
    sematt_kernel<<<NVD, 128, 0, stream>>>(out_tv, sa_w1, sa_b1, sa_w2, wacc + 0);
    sematt_kernel<<<NVD, 128, 0, stream>>>(out_vv, sa_w1, sa_b1, sa_w2, wacc + 1);
    beta_kernel<<<1, 1, 0, stream>>>(wacc, beta, 1.0f / (float)NVD);

    final_video_kernel<<<NVD, 128, 0, stream>>>(out_tv, out_vv, beta, out);
    copy_kernel<<<(NTD * HIDDEN + 255) / 256, 256, 0, stream>>>(
        out_vt, out + (size_t)NVD * HIDDEN, NTD * HIDDEN);
}